// WeightedMyLoss_69483980914852
// MI455X (gfx1250) — compile-verified
//
#include <hip/hip_runtime.h>
#include <cstdint>

#define BDIM 8
#define CDIM 19
#define HDIM 512
#define WDIM 1024
#define TW 64
#define TH 16
#define HALO 7
#define TPB 256
#define TILES_X (WDIM / TW)                 // 16
#define TILES_Y (HDIM / TH)                 // 32
#define TILES_PER_IMG (TILES_X * TILES_Y)   // 512
#define NBLOCKS (BDIM * TILES_PER_IMG)      // 4096
#define LDS_W (TW + 2 * HALO)               // 78
#define LDS_H (TH + 2 * HALO)               // 30

typedef unsigned int u32x4 __attribute__((ext_vector_type(4)));
typedef int          i32x8 __attribute__((ext_vector_type(8)));
typedef int          i32x4 __attribute__((ext_vector_type(4)));
typedef float        f32x4 __attribute__((ext_vector_type(4)));

__global__ void __launch_bounds__(TPB)
fused_ce_boundary_kernel(const float* __restrict__ logits,
                         const int*   __restrict__ targets,
                         float*       __restrict__ partials)
{
    __shared__ int   tile[LDS_W * LDS_H];
    __shared__ float redP[TPB];
    __shared__ float redS[TPB];

    const int blk = blockIdx.x;
    const int b   = blk / TILES_PER_IMG;
    const int t   = blk % TILES_PER_IMG;
    const int ty  = t / TILES_X;
    const int tx  = t % TILES_X;
    const int w0  = tx * TW;
    const int h0  = ty * TH;

    // Clamped (halo'd) tile footprint actually present in the image.
    const int gx0 = max(w0 - HALO, 0);
    const int gx1 = min(w0 + TW + HALO, WDIM);
    const int gy0 = max(h0 - HALO, 0);
    const int gy1 = min(h0 + TH + HALO, HDIM);
    const int gw  = gx1 - gx0;   // LDS row stride (elements)
    const int gh  = gy1 - gy0;

    // ---- Issue TDM async tensor load of the target tile (wave 0 only).
    // TDM ignores EXEC, so the gate must be a scalar branch: readfirstlane
    // of the wave id forces an SGPR condition.
    if (__builtin_amdgcn_readfirstlane((int)(threadIdx.x >> 5)) == 0) {
        const uint64_t gaddr =
            (uint64_t)(uintptr_t)(targets + ((size_t)b * HDIM + gy0) * WDIM + gx0);
        const uint32_t laddr = (uint32_t)(uintptr_t)(&tile[0]);

        // D# Group 0 (128b): count=1 | lds_addr | global_addr[56:0] | type=2
        u32x4 g0;
        g0.x = 1u;
        g0.y = laddr;
        g0.z = (uint32_t)gaddr;
        g0.w = (uint32_t)((gaddr >> 32) & 0x01FFFFFFu) | 0x80000000u;

        // D# Group 1 (256b): data_size=4B, tensor_dim0/1, tile_dim0/1,
        // tensor_dim0_stride = WDIM (row pitch of the label image).
        i32x8 g1;
        g1[0] = 0x00020000;                                      // data_size=2 (4B)
        g1[1] = (gw & 0xFFFF) << 16;                             // tensor_dim0 lo
        g1[2] = ((gw >> 16) & 0xFFFF) | ((gh & 0xFFFF) << 16);   // td0 hi | td1 lo
        g1[3] = ((gh >> 16) & 0xFFFF) | ((gw & 0xFFFF) << 16);   // td1 hi | tile_dim0
        g1[4] = (gh & 0xFFFF);                                   // tile_dim1 (tile_dim2=0)
        g1[5] = WDIM;                                            // dim0_stride lo32
        g1[6] = 0;                                               // dim0_stride hi | dim1_stride lo
        g1[7] = 0;

        const i32x4 gz4 = {0, 0, 0, 0};                          // groups 2/3 unused (2D)
        const i32x8 gz8 = {0, 0, 0, 0, 0, 0, 0, 0};
        __builtin_amdgcn_tensor_load_to_lds(g0, g1, gz4, gz4, gz8, 0);
    }

    // ---- Thread layout: 1 row, 4 consecutive columns -> 16B vector loads.
    const int lw = (int)(threadIdx.x % 16) * 4;   // column base within tile
    const int lh = (int)(threadIdx.x / 16);       // row within tile
    const int w  = w0 + lw;
    const int h  = h0 + lh;

    // One 16B load for the 4 target labels of this thread.
    const i32x4 tg4 = *reinterpret_cast<const i32x4*>(
        targets + ((size_t)b * HDIM + h) * WDIM + w);
    const int tg[4] = {tg4.x, tg4.y, tg4.z, tg4.w};

    float m[4], ssum[4], xt[4];
#pragma unroll
    for (int p = 0; p < 4; ++p) {
        m[p]    = -3.402823466e+38f;
        ssum[p] = 0.0f;
        xt[p]   = 0.0f;
    }

    // ---- Online log-sum-exp over 19 channels while the DMA is in flight.
    // Non-temporal: the 319 MB logit stream is read exactly once.
    const size_t pixbase = ((size_t)b * CDIM * HDIM + h) * WDIM + w;
    for (int c = 0; c < CDIM; ++c) {
        const f32x4 x4 = __builtin_nontemporal_load(
            reinterpret_cast<const f32x4*>(logits + pixbase + (size_t)c * HDIM * WDIM));
        const float xs[4] = {x4.x, x4.y, x4.z, x4.w};
#pragma unroll
        for (int p = 0; p < 4; ++p) {
            const float x  = xs[p];
            const float mn = fmaxf(m[p], x);
            ssum[p] = ssum[p] * __expf(m[p] - mn) + __expf(x - mn);
            m[p]    = mn;
            if (c == tg[p]) xt[p] = x;
        }
    }

    __builtin_amdgcn_s_wait_tensorcnt(0);
    __syncthreads();

    // ---- Cross morphology (15-arm dilate/erode) from LDS + weighted loss.
    const int ylo = max(h - HALO, 0), yhi = min(h + HALO, HDIM - 1);
    float locP = 0.0f, locS = 0.0f;
#pragma unroll
    for (int p = 0; p < 4; ++p) {
        const int wp  = w + p;
        const int xlo = max(wp - HALO, 0), xhi = min(wp + HALO, WDIM - 1);

        int vmx = tile[(h - gy0) * gw + (wp - gx0)];
        int vmn = vmx;
        const int rowbase = (h - gy0) * gw - gx0;
        for (int x = xlo; x <= xhi; ++x) {
            const int v = tile[rowbase + x];
            vmx = max(vmx, v); vmn = min(vmn, v);
        }
        const int colbase = (wp - gx0) - gy0 * gw;
        for (int y = ylo; y <= yhi; ++y) {
            const int v = tile[colbase + y * gw];
            vmx = max(vmx, v); vmn = min(vmn, v);
        }
        const float wt   = (float)((vmx ^ vmn) & 0xFF) + 1.0f;   // uint8 xor + 1
        const float loss = __logf(ssum[p]) + m[p] - xt[p];
        locP += loss * wt;
        locS += wt;
    }

    // ---- Deterministic block reduction -> per-block partials.
    redP[threadIdx.x] = locP;
    redS[threadIdx.x] = locS;
    __syncthreads();
    for (int s = TPB >> 1; s > 0; s >>= 1) {
        if ((int)threadIdx.x < s) {
            redP[threadIdx.x] += redP[threadIdx.x + s];
            redS[threadIdx.x] += redS[threadIdx.x + s];
        }
        __syncthreads();
    }
    if (threadIdx.x == 0) {
        partials[2 * blk + 0] = redP[0];
        partials[2 * blk + 1] = redS[0];
    }
}

__global__ void __launch_bounds__(TPB)
finalize_kernel(const float* __restrict__ partials, float* __restrict__ out)
{
    __shared__ float redP[TPB];
    __shared__ float redS[TPB];

    float total = 0.0f;
    for (int b = 0; b < BDIM; ++b) {
        float p = 0.0f, s = 0.0f;
        for (int i = (int)threadIdx.x; i < TILES_PER_IMG; i += TPB) {
            p += partials[2 * (b * TILES_PER_IMG + i) + 0];
            s += partials[2 * (b * TILES_PER_IMG + i) + 1];
        }
        redP[threadIdx.x] = p;
        redS[threadIdx.x] = s;
        __syncthreads();
        for (int k = TPB >> 1; k > 0; k >>= 1) {
            if ((int)threadIdx.x < k) {
                redP[threadIdx.x] += redP[threadIdx.x + k];
                redS[threadIdx.x] += redS[threadIdx.x + k];
            }
            __syncthreads();
        }
        if (threadIdx.x == 0) total += redP[0] / redS[0];
        __syncthreads();
    }
    // sum(loss*w_norm)/sum(w_norm) = (1/B) * sum_b P_b / S_b
    if (threadIdx.x == 0) out[0] = total / (float)BDIM;
}

extern "C" void kernel_launch(void* const* d_in, const int* in_sizes, int n_in,
                              void* d_out, int out_size, void* d_ws, size_t ws_size,
                              hipStream_t stream) {
    const float* logits   = (const float*)d_in[0];
    const int*   targets  = (const int*)d_in[1];
    float*       out      = (float*)d_out;
    float*       partials = (float*)d_ws;   // NBLOCKS * 2 floats = 32 KB

    fused_ce_boundary_kernel<<<NBLOCKS, TPB, 0, stream>>>(logits, targets, partials);
    finalize_kernel<<<1, TPB, 0, stream>>>(partials, out);
}